// UUIIModel_14456859918736
// MI455X (gfx1250) — compile-verified
//
#include <hip/hip_runtime.h>

// Problem constants (from reference).
#define NU_N 100000
#define NI_N 150000
#define KDIM 64
#define TK 20
#define ROWS_PER_BLOCK 8   // 256 threads = 8 wave32 per block, 1 row per wave

typedef float __attribute__((ext_vector_type(2))) f2;

// -------- CDNA5 async-to-LDS path (gfx1250 only, builtin-guarded) ----------
#if defined(__gfx1250__) && __has_builtin(__builtin_amdgcn_global_load_async_to_lds_b128)
#define USE_ASYNC_LDS 1
#else
#define USE_ASYNC_LDS 0
#endif

#if USE_ASYNC_LDS && __has_builtin(__builtin_amdgcn_s_wait_asynccnt)
#define HAVE_WAIT_ASYNC 1
#else
#define HAVE_WAIT_ASYNC 0
#endif

// Uniform-lane broadcast via v_readlane (SALU/SGPR path, no ds_bpermute).
__device__ __forceinline__ int readlane_i(int v, int l) {
  return __builtin_amdgcn_readlane(v, l);
}
__device__ __forceinline__ float readlane_f(float v, int l) {
  return __int_as_float(__builtin_amdgcn_readlane(__float_as_int(v), l));
}

#if USE_ASYNC_LDS
// The builtin wants 128-bit-vector-typed pointers: AS1 (global) source and
// AS3 (LDS) destination.
typedef int v4i __attribute__((vector_size(16)));
typedef __attribute__((address_space(1))) v4i as1_v4i;
typedef __attribute__((address_space(3))) v4i as3_v4i;

// Generic->AS1: flat global address == AS1 address (64-bit). Route through an
// integer so no cross-address-space pointer cast is needed.
__device__ __forceinline__ as1_v4i* to_global(const void* p) {
  return (as1_v4i*)(unsigned long long)p;
}
// Generic->AS3: low 32 bits of a flat LDS pointer are the LDS byte address
// AS3 pointers are 32-bit.
__device__ __forceinline__ as3_v4i* to_lds(const void* p) {
  return (as3_v4i*)(unsigned)(unsigned long long)p;
}
#endif

#if HAVE_WAIT_ASYNC
// Pipelined consume: async loads complete in order, so before consuming
// neighbor pair K we only need asynccnt <= (TK/2-1-K). Template recursion
// makes the wait immediate a true constant expression (builtin requirement).
template <int K>
__device__ __forceinline__ void pipelined_accum(const float* wl, int lane,
                                                float wt, f2& acc) {
  if constexpr (K < TK / 2) {
    __builtin_amdgcn_s_wait_asynccnt(TK / 2 - 1 - K);  // issue K done
#pragma unroll
    for (int j = 0; j < 2; ++j) {
      const int nb = 2 * K + j;
      const float w = readlane_f(wt, nb);              // scalar operand
      f2 v = *(const f2*)(wl + nb * KDIM + lane * 2);  // conflict-free 8B/lane
      acc += w * v;
    }
    pipelined_accum<K + 1>(wl, lane, wt, acc);
  }
}
#endif

// ---------------------------------------------------------------------------
// Kernel 1: dinv[r] = rowsum > 0 ? rsqrt(max(rowsum, 1e-30)) : 0
// Row r owns vals[r*TK .. r*TK+19] (rows = repeat(arange(n), TK)).
// ---------------------------------------------------------------------------
__global__ void deg_rsqrt_kernel(const float* __restrict__ vals,
                                 float* __restrict__ dinv, int n) {
  int r = blockIdx.x * blockDim.x + threadIdx.x;
  if (r >= n) return;
  const float* v = vals + (long long)r * TK;
  float s = 0.f;
#pragma unroll
  for (int k = 0; k < TK; ++k) s += v[k];
  dinv[r] = (s > 0.f) ? __frsqrt_rn(fmaxf(s, 1e-30f)) : 0.f;
}

// ---------------------------------------------------------------------------
// Kernel 2: one wave per row.
//   acc[row,:] = sum_k w[row,k] * x[col[row,k], :]          (SpMM)
// FINAL=false:  y = acc                                     (x1 pass)
// FINAL=true :  y = (g + x + acc) / 3                       (x2 + mean pass)
// Gather path: GLOBAL_LOAD_ASYNC_TO_LDS_B128 (2 neighbor rows / instruction,
// 10 instructions / row) -> pipelined s_wait_asynccnt -> LDS accumulate.
// ---------------------------------------------------------------------------
template <bool FINAL>
__global__ __launch_bounds__(256) void spmm_kernel(
    const float* __restrict__ x, const float* __restrict__ g,
    const float* __restrict__ dinv, const int* __restrict__ cols,
    const float* __restrict__ vals, float* __restrict__ y, int n) {
  const int lane = threadIdx.x & 31;
  const int wv = threadIdx.x >> 5;
  const int row = blockIdx.x * ROWS_PER_BLOCK + wv;
#if USE_ASYNC_LDS
  __shared__ __align__(16) float smem[ROWS_PER_BLOCK * TK * KDIM];  // 40 KB
#endif
  if (row >= n) return;  // whole wave uniform

  const long long base = (long long)row * TK;
  const float dr = dinv[row];

  // Lanes 0..19 stage (col, weight) for this row's 20 edges.
  int c = 0;
  float wt = 0.f;
  if (lane < TK) {
    c = cols[base + lane];
    wt = vals[base + lane] * dr * dinv[c];
  }

  f2 acc = {0.f, 0.f};

#if USE_ASYNC_LDS
  float* wl = smem + wv * (TK * KDIM);
  const int sub = lane & 15;      // 16 lanes x 16B = one 256B neighbor row
  const bool hi = (lane >> 4);    // upper half-wave handles odd neighbor
#pragma unroll
  for (int k = 0; k < TK / 2; ++k) {
    // Neighbor pair (2k, 2k+1): broadcast both cols via readlane, select.
    const int c0 = readlane_i(c, 2 * k);
    const int c1 = readlane_i(c, 2 * k + 1);
    const int cn = hi ? c1 : c0;
    const int nb = 2 * k + (hi ? 1 : 0);
    const float* src = x + ((long long)cn << 6) + sub * 4;
    float* dst = wl + nb * KDIM + sub * 4;
    __builtin_amdgcn_global_load_async_to_lds_b128(to_global(src), to_lds(dst),
                                                   0, 0);
  }
#if HAVE_WAIT_ASYNC
  pipelined_accum<0>(wl, lane, wt, acc);
#else
  asm volatile("s_wait_asynccnt 0" ::: "memory");
#pragma unroll
  for (int k = 0; k < TK; ++k) {
    const float w = readlane_f(wt, k);
    f2 v = *(const f2*)(wl + k * KDIM + lane * 2);
    acc += w * v;
  }
#endif
#else
  // Fallback: direct coalesced gather (256B per neighbor row across the wave).
#pragma unroll
  for (int k = 0; k < TK; ++k) {
    const int ck = readlane_i(c, k);
    const float w = readlane_f(wt, k);
    f2 v = *(const f2*)(x + ((long long)ck << 6) + lane * 2);
    acc += w * v;
  }
#endif

  float* yp = y + ((long long)row << 6) + lane * 2;
  if (FINAL) {
    f2 gv = *(const f2*)(g + ((long long)row << 6) + lane * 2);
    f2 xv = *(const f2*)(x + ((long long)row << 6) + lane * 2);
    f2 res = (gv + xv + acc) * (1.0f / 3.0f);
    __builtin_nontemporal_store(res, (f2*)yp);  // streamed output, don't cache
  } else {
    *(f2*)yp = acc;  // x1: re-read next pass, keep cacheable
  }
}

// ---------------------------------------------------------------------------
// Launch: UU graph then II graph, workspace reused.
//   ws layout: [ dinv : 150016 floats ][ x1 : 150000*64 floats ]  (~39 MB)
//   out layout: [ gu_out : NU*64 ][ gi_out : NI*64 ]
// ---------------------------------------------------------------------------
extern "C" void kernel_launch(void* const* d_in, const int* in_sizes, int n_in,
                              void* d_out, int out_size, void* d_ws,
                              size_t ws_size, hipStream_t stream) {
  const float* gu = (const float*)d_in[0];
  const float* gi = (const float*)d_in[1];
  const float* uu_val = (const float*)d_in[2];
  const float* ii_val = (const float*)d_in[3];
  const int* uu_col = (const int*)d_in[4];
  const int* ii_col = (const int*)d_in[5];

  float* out_u = (float*)d_out;
  float* out_i = out_u + (long long)NU_N * KDIM;

  float* dinv = (float*)d_ws;
  float* x1 = dinv + 150016;  // 16B-aligned

  const dim3 blk(256);
  const dim3 grid_deg_u((NU_N + 255) / 256), grid_deg_i((NI_N + 255) / 256);
  const dim3 grid_u((NU_N + ROWS_PER_BLOCK - 1) / ROWS_PER_BLOCK);
  const dim3 grid_i((NI_N + ROWS_PER_BLOCK - 1) / ROWS_PER_BLOCK);

  // ---- UU graph ----
  deg_rsqrt_kernel<<<grid_deg_u, blk, 0, stream>>>(uu_val, dinv, NU_N);
  spmm_kernel<false><<<grid_u, blk, 0, stream>>>(gu, nullptr, dinv, uu_col,
                                                 uu_val, x1, NU_N);
  spmm_kernel<true><<<grid_u, blk, 0, stream>>>(x1, gu, dinv, uu_col, uu_val,
                                                out_u, NU_N);
  // ---- II graph (reuses dinv/x1; stream ordering guarantees safety) ----
  deg_rsqrt_kernel<<<grid_deg_i, blk, 0, stream>>>(ii_val, dinv, NI_N);
  spmm_kernel<false><<<grid_i, blk, 0, stream>>>(gi, nullptr, dinv, ii_col,
                                                 ii_val, x1, NI_N);
  spmm_kernel<true><<<grid_i, blk, 0, stream>>>(x1, gi, dinv, ii_col, ii_val,
                                                out_i, NI_N);
}